// AttentionDist_42279658062534
// MI455X (gfx1250) — compile-verified
//
#include <hip/hip_runtime.h>
#include <hip/hip_bf16.h>
#include <math.h>
#include <stdint.h>

// Problem constants (fixed by the reference setup_inputs)
#define BN   8
#define QN   1024
#define KN   4096
#define DN   64
#define QT   16          // q-rows per workgroup
#define SSTR 4100        // padded LDS stride for the logits row-block (4096+4)
#define QSTR 68          // padded LDS stride for the Q tile (64+4)
#define MAXC 32          // >= num_classes+1 (=21)

typedef __attribute__((ext_vector_type(2))) float v2f;
typedef __attribute__((ext_vector_type(8))) float v8f;
typedef __attribute__((ext_vector_type(4))) unsigned int u32x4;
typedef __attribute__((ext_vector_type(4))) int i32x4;
typedef __attribute__((ext_vector_type(8))) int i32x8;

#if defined(__HIP_DEVICE_COMPILE__) && __has_builtin(__builtin_amdgcn_tensor_load_to_lds)
#define HAVE_TDM 1
#else
#define HAVE_TDM 0
#endif

static __device__ __forceinline__ v8f wmma_f32x4(v2f a, v2f b, v8f c) {
  // D = A(16x4,f32) * B(4x16,f32) + C(16x16,f32)
  return __builtin_amdgcn_wmma_f32_16x16x4_f32(false, a, false, b, (short)0, c,
                                               false, false);
}

// float atomic-max on LDS via int/uint ordering trick (works for all finite
// floats and -inf init; positive floats ordered as ints, negative as uints)
static __device__ __forceinline__ void atomicMaxF(float* addr, float v) {
  if (v >= 0.0f)
    atomicMax((int*)addr, __float_as_int(v));
  else
    atomicMin((unsigned int*)addr, __float_as_uint(v));
}

#if HAVE_TDM
// Tensor Data Mover: 2D tile of 32-bit elements, global -> LDS.
//   rows x cols dwords, row stride `stride` dwords, optional LDS row padding:
//   pad_interval_code: 0=2,1=4,...,5=64,...  pad_amount_code: 0=1,...,3=4 dwords
static __device__ __forceinline__ void
tdm_load_2d_b32(unsigned lds_byte_off, const void* gptr, unsigned cols,
                unsigned rows, unsigned stride, int pad_en,
                unsigned pad_interval_code, unsigned pad_amount_code) {
  uint64_t ga = (uint64_t)(uintptr_t)gptr;
  u32x4 g0;
  g0[0] = 1u;                                   // count=1, is_restore=0
  g0[1] = lds_byte_off;                         // lds_addr (bytes)
  g0[2] = (unsigned)(ga & 0xFFFFFFFFu);         // global_addr[31:0]
  g0[3] = (unsigned)((ga >> 32) & 0x01FFFFFFu)  // global_addr[56:32]
        | (2u << 30);                           // type = 2 ("image")
  i32x8 g1;
  g1[0] = (int)((2u << 16)                      // data_size = 4 bytes
        | ((unsigned)(pad_en ? 1u : 0u) << 20)
        | (pad_interval_code << 22)
        | (pad_amount_code << 25));
  g1[1] = (int)((cols & 0xFFFFu) << 16);        // tensor_dim0[15:0]
  g1[2] = (int)(((cols >> 16) & 0xFFFFu)        // tensor_dim0[31:16]
        | ((rows & 0xFFFFu) << 16));            // tensor_dim1[15:0]
  g1[3] = (int)(((rows >> 16) & 0xFFFFu)        // tensor_dim1[31:16]
        | ((cols & 0xFFFFu) << 16));            // tile_dim0
  g1[4] = (int)(rows & 0xFFFFu);                // tile_dim1 (tile_dim2=0)
  g1[5] = (int)stride;                          // tensor_dim0_stride[31:0]
  g1[6] = 0;                                    // stride hi / dim1_stride lo
  g1[7] = 0;
  i32x4 z4 = {0, 0, 0, 0};
#if __clang_major__ >= 23
  i32x8 z8 = {0, 0, 0, 0, 0, 0, 0, 0};
  __builtin_amdgcn_tensor_load_to_lds(g0, g1, z4, z4, z8, 0);
#else
  __builtin_amdgcn_tensor_load_to_lds(g0, g1, z4, z4, 0);
#endif
}
#endif

__global__ __launch_bounds__(256) void
attn_dist_kernel(const float* __restrict__ qry, const float* __restrict__ ctx,
                 const int* __restrict__ labels, const float* __restrict__ mask,
                 const int* __restrict__ ncls, float* __restrict__ out) {
  extern __shared__ float smem[];
  float* sS      = smem;                       // [QT][SSTR] logits -> P
  float* sQ      = sS + QT * SSTR;             // [QT][QSTR]
  int*   sLab    = (int*)(sQ + QT * QSTR);     // [KN]
  float* sOp     = (float*)(sLab + KN);        // [8][256] partial O tiles
  float* sCmax   = sOp + 8 * 256;              // [QT][MAXC]
  float* sCsum   = sCmax + QT * MAXC;          // [QT][MAXC]
  float* sRowMax = sCsum + QT * MAXC;          // [QT]
  float* sRowSum = sRowMax + QT;               // [QT]

  const int tid  = threadIdx.x;
  const int lane = tid & 31;
  const int wave = tid >> 5;
  const int lo16 = lane & 15;
  const int hi   = lane >> 4;       // 0 or 1

  const int bIdx = blockIdx.x / (QN / QT);
  const int q0   = (blockIdx.x % (QN / QT)) * QT;
  const int C1   = ncls[0] + 1;

  const float NEG_INF = __int_as_float(0xFF800000);

  // ---- stage Q tile + labels via the Tensor Data Mover ---------------------
#if HAVE_TDM
  if (wave == 0) {
    // Q tile: 16 rows x 64 dwords, LDS rows padded 64 -> 68 dwords
    // (pad_interval code 5 = 64 dwords, pad_amount code 3 = 4 dwords)
    tdm_load_2d_b32((unsigned)((QT * SSTR) * 4),
                    qry + (size_t)(bIdx * QN + q0) * DN,
                    DN, QT, DN, /*pad=*/1, 5u, 3u);
    // Labels: flat 4096 x 1 dwords
    tdm_load_2d_b32((unsigned)((QT * SSTR + QT * QSTR) * 4),
                    labels + (size_t)bIdx * KN,
                    KN, 1, KN, /*pad=*/0, 0u, 0u);
    __builtin_amdgcn_s_wait_tensorcnt(0);
  }
#else
  for (int t = tid; t < QT * DN; t += 256) {
    int r = t >> 6, c = t & 63;
    sQ[r * QSTR + c] = qry[((size_t)(bIdx * QN + q0 + r)) * DN + c];
  }
  for (int t = tid; t < KN; t += 256) sLab[t] = labels[bIdx * KN + t];
#endif
  for (int t = tid; t < QT * MAXC; t += 256) {
    sCmax[t] = NEG_INF;
    sCsum[t] = 0.0f;
  }
  if (tid < QT) sRowSum[tid] = 0.0f;
  __syncthreads();

  // ---- Phase 1: logits row-block via V_WMMA_F32_16X16X4_F32 ----------------
  // Each wave computes 16x16 tiles of S = Qtile(16x64) x Ctx^T, chained over D.
  const float* ctxB = ctx + (size_t)bIdx * KN * DN;
  for (int chunk = wave; chunk < KN / 16; chunk += 8) {
    const int k0 = chunk * 16;
    v8f acc = {};
#pragma unroll
    for (int d = 0; d < DN; d += 4) {
      v2f av = *(const v2f*)&sQ[lo16 * QSTR + d + 2 * hi];
      v2f bv = *(const v2f*)&ctxB[(size_t)(k0 + lo16) * DN + d + 2 * hi];
      acc = wmma_f32x4(av, bv, acc);
    }
    const float mval = mask[bIdx * KN + k0 + lo16];
#pragma unroll
    for (int r = 0; r < 8; ++r) {
      int M = r + 8 * hi;
      sS[M * SSTR + k0 + lo16] = acc[r] * mval;
    }
  }
  __syncthreads();

  // ---- Phase 2a: per-class max (and hence row max) -------------------------
  {
    const int row = tid >> 4, l16 = tid & 15;
    for (int k = l16; k < KN; k += 16) {
      float x = sS[row * SSTR + k];
      int c = sLab[k];
      atomicMaxF(&sCmax[row * MAXC + c], x);
    }
  }
  __syncthreads();

  // ---- Phase 2b: row max = max over classes --------------------------------
  if (tid < QT) {
    float m = NEG_INF;
    for (int c = 0; c < C1; ++c) m = fmaxf(m, sCmax[tid * MAXC + c]);
    sRowMax[tid] = m;
  }
  __syncthreads();

  // ---- Phase 2c: exp pass: P = exp(S - rowmax), row sums, class sums -------
  {
    const int row = tid >> 4, l16 = tid & 15;
    const float rm = sRowMax[row];
    float lsum = 0.0f;
    for (int k = l16; k < KN; k += 16) {
      float x = sS[row * SSTR + k];
      float p = __expf(x - rm);
      sS[row * SSTR + k] = p;                 // overwrite with P
      lsum += p;
      int c = sLab[k];
      atomicAdd(&sCsum[row * MAXC + c], __expf(x - sCmax[row * MAXC + c]));
    }
    atomicAdd(&sRowSum[row], lsum);
  }
  __syncthreads();

  // ---- Phase 3: O = P(16xK) x Ctx(KxD) via WMMA ----------------------------
  // 8 waves: wave&3 -> 16-wide d-tile, wave>>2 -> K half.
  {
    const int d0 = (wave & 3) * 16;
    const int kb = (wave >> 2) * (KN / 2);
    v8f acc = {};
#pragma unroll 4
    for (int k = kb; k < kb + KN / 2; k += 4) {
      v2f av = *(const v2f*)&sS[lo16 * SSTR + k + 2 * hi];
      v2f bv;
      bv.x = ctxB[(size_t)(k + 2 * hi) * DN + d0 + lo16];
      bv.y = ctxB[(size_t)(k + 2 * hi + 1) * DN + d0 + lo16];
      acc = wmma_f32x4(av, bv, acc);
    }
#pragma unroll
    for (int r = 0; r < 8; ++r) {
      int M = r + 8 * hi;
      sOp[wave * 256 + M * 16 + lo16] = acc[r];
    }
  }
  __syncthreads();

  // ---- Phase 4: combine halves, normalize, store context_vector ------------
  float* outCV  = out;
  float* outCls = out + (size_t)BN * QN * DN;
  for (int t = tid; t < QT * DN; t += 256) {
    int M = t >> 6, d = t & 63;
    int dt = d >> 4, N = d & 15;
    float v = sOp[dt * 256 + M * 16 + N] + sOp[(dt + 4) * 256 + M * 16 + N];
    v /= sRowSum[M];
    outCV[((size_t)(bIdx * QN + q0 + M)) * DN + d] = v;
  }

  // ---- Phase 5: grouped logsumexp output -----------------------------------
  for (int t = tid; t < QT * C1; t += 256) {
    int row = t / C1, c = t % C1;
    float lse = sRowMax[row] + __logf(sRowSum[row]);
    float v = sCmax[row * MAXC + c] + __logf(sCsum[row * MAXC + c]) - lse;
    outCls[((size_t)(bIdx * QN + q0 + row)) * C1 + c] = v;
  }
}

extern "C" void kernel_launch(void* const* d_in, const int* in_sizes, int n_in,
                              void* d_out, int out_size, void* d_ws, size_t ws_size,
                              hipStream_t stream) {
  (void)in_sizes; (void)n_in; (void)d_ws; (void)ws_size; (void)out_size;
  const float* qry    = (const float*)d_in[0];
  const float* ctx    = (const float*)d_in[1];
  const int*   labels = (const int*)d_in[2];
  const float* mask   = (const float*)d_in[3];
  const int*   ncls   = (const int*)d_in[4];
  float*       out    = (float*)d_out;

  constexpr size_t smemWords = (size_t)QT * SSTR      // sS
                             + (size_t)QT * QSTR      // sQ
                             + (size_t)KN             // sLab
                             + 8 * 256                // sOp
                             + 2 * QT * MAXC          // sCmax, sCsum
                             + 2 * QT;                // sRowMax, sRowSum
  constexpr size_t smemBytes = smemWords * 4;

  static_assert(smemBytes <= 320u * 1024u, "LDS over WGP budget");
  (void)hipFuncSetAttribute((const void*)attn_dist_kernel,
                            hipFuncAttributeMaxDynamicSharedMemorySize,
                            (int)smemBytes);

  dim3 grid(BN * (QN / QT));
  dim3 block(256);
  attn_dist_kernel<<<grid, block, smemBytes, stream>>>(qry, ctx, labels, mask,
                                                       ncls, out);
}